// Neural_Network_56092272886487
// MI455X (gfx1250) — compile-verified
//
#include <hip/hip_runtime.h>

typedef __attribute__((ext_vector_type(2))) float v2f;
typedef __attribute__((ext_vector_type(8))) float v8f;

namespace {
constexpr int NI = 5;          // input features
constexpr int NH = 32;         // hidden width
constexpr int H1S = 33;        // padded LDS row stride (gcd(33,64)=1 -> no bank conflicts)
constexpr int WAVES = 8;       // 256 threads / wave32
}

__device__ __forceinline__ float fast_tanh(float x) {
#if __has_builtin(__builtin_amdgcn_tanhf)
    return __builtin_amdgcn_tanhf(x);          // v_tanh_f32 on gfx1250
#else
    // tanh(x) = 1 - 2/(exp(2x)+1); __expf lowers to v_exp_f32. Robust at +-inf.
    float e = __expf(2.0f * x);
    return 1.0f - 2.0f / (e + 1.0f);
#endif
}

__global__ __launch_bounds__(256) void grouped_mlp_wmma(
    const float* __restrict__ g,  const float* __restrict__ W1,
    const float* __restrict__ b1, const float* __restrict__ W2,
    const float* __restrict__ b2, const float* __restrict__ W3,
    const float* __restrict__ b3, float* __restrict__ out,
    int S, int A)
{
    __shared__ float W1s[NI * NH];
    __shared__ float b1s[NH];
    __shared__ float W2s[NH * NH];       // [h_in][h_out], row-major as in memory
    __shared__ float b2s[NH];
    __shared__ float W3s[NH];
    __shared__ float b3s;
    __shared__ float h1s[WAVES][16 * H1S];  // per-wave 16x32 h1 tile, padded

    const int atom = blockIdx.x;
    const int tid  = threadIdx.x;
    const int lane = tid & 31;
    const int wave = tid >> 5;

    // ---- stage this atom's weights into LDS ----
    for (int i = tid; i < NI * NH; i += 256) W1s[i] = W1[(size_t)atom * NI * NH + i];
    for (int i = tid; i < NH * NH; i += 256) W2s[i] = W2[(size_t)atom * NH * NH + i];
    if (tid < NH) {
        b1s[tid] = b1[(size_t)atom * NH + tid];
        b2s[tid] = b2[(size_t)atom * NH + tid];
        W3s[tid] = W3[(size_t)atom * NH + tid];
    }
    if (tid == 0) b3s = b3[atom];
    __syncthreads();

    const int m    = lane & 15;   // struct-in-tile (A rows) / N column index
    const int half = lane >> 4;
    float* h1w = &h1s[wave][0];

    const size_t gstride = (size_t)A * NI;   // floats between structures in g
    const int ntiles = (S + 15) >> 4;        // 125 for S=2000

    // Each wave owns tiles t = wave, wave+8, ... : trip count uniform per wave,
    // so EXEC is all-ones around every WMMA.
    for (int t = wave; t < ntiles; t += WAVES) {
        const int s0 = t * 16;
        int srow = s0 + m; if (srow > S - 1) srow = S - 1;   // safe for ragged S

        // prefetch next tile's g rows (global_prefetch_b8)
        if (t + WAVES < ntiles) {
            const float* pf = g + (size_t)(s0 + 16 * WAVES + m) * gstride + (size_t)atom * NI;
            __builtin_prefetch(pf, 0, 3);
        }

        // ---- layer 1: h1 = tanh(g @ W1 + b1), K=5 on VALU ----
        const float* gp = g + (size_t)srow * gstride + (size_t)atom * NI;
        float g0 = gp[0], g1 = gp[1], g2 = gp[2], g3 = gp[3], g4 = gp[4];
        #pragma unroll
        for (int j = 0; j < 16; ++j) {
            const int h = half * 16 + j;
            float acc = b1s[h];
            acc = fmaf(g0, W1s[0 * NH + h], acc);
            acc = fmaf(g1, W1s[1 * NH + h], acc);
            acc = fmaf(g2, W1s[2 * NH + h], acc);
            acc = fmaf(g3, W1s[3 * NH + h], acc);
            acc = fmaf(g4, W1s[4 * NH + h], acc);
            h1w[m * H1S + h] = fast_tanh(acc);
        }
        // wave-private LDS region; same-wave DS ops are in-order -> no barrier

        // ---- layer 2: (16x32) @ (32x32) via 8 chained v_wmma_f32_16x16x4_f32
        //      per N-halftile; accumulator pre-seeded with b2 (bias over N) ----
        const float bias2a = b2s[m];
        const float bias2b = b2s[m + 16];
        v8f acc0, acc1;
        #pragma unroll
        for (int v = 0; v < 8; ++v) { acc0[v] = bias2a; acc1[v] = bias2b; }

        #pragma unroll
        for (int kk = 0; kk < 8; ++kk) {
            // A 16x4 f32 layout: lanes 0-15 hold K={4kk,4kk+1}, lanes 16-31 K={4kk+2,4kk+3}
            const int k0 = kk * 4 + half * 2;
            v2f a, bb0, bb1;
            a[0] = h1w[m * H1S + k0];
            a[1] = h1w[m * H1S + k0 + 1];
            // B 4x16 f32: N across lanes, same K split across lane halves
            bb0[0] = W2s[(k0    ) * NH + m];
            bb0[1] = W2s[(k0 + 1) * NH + m];
            bb1[0] = W2s[(k0    ) * NH + m + 16];
            bb1[1] = W2s[(k0 + 1) * NH + m + 16];
            acc0 = __builtin_amdgcn_wmma_f32_16x16x4_f32(
                       false, a, false, bb0, (short)0, acc0, false, false);
            acc1 = __builtin_amdgcn_wmma_f32_16x16x4_f32(
                       false, a, false, bb1, (short)0, acc1, false, false);
        }

        // ---- tanh + layer 3: e = h2 @ W3 + b3 (reduce over N across lanes) ----
        const float w3a = W3s[m];
        const float w3b = W3s[m + 16];
        float ev[8];
        #pragma unroll
        for (int v = 0; v < 8; ++v) {
            const float h2a = fast_tanh(acc0[v]);
            const float h2b = fast_tanh(acc1[v]);
            float p = fmaf(h2a, w3a, h2b * w3b);
            // butterfly reduce within 16-lane halves (C/D: lanes0-15 M=v, 16-31 M=v+8)
            p += __shfl_xor(p, 1, 32);
            p += __shfl_xor(p, 2, 32);
            p += __shfl_xor(p, 4, 32);
            p += __shfl_xor(p, 8, 32);
            ev[v] = p + b3s;
        }
        if (m == 0) {   // lane 0 writes M=0..7, lane 16 writes M=8..15
            const int sbase = s0 + half * 8;
            #pragma unroll
            for (int v = 0; v < 8; ++v) {
                const int s = sbase + v;
                if (s < S) out[(size_t)s * A + atom] = ev[v];
            }
        }
    }
}

extern "C" void kernel_launch(void* const* d_in, const int* in_sizes, int n_in,
                              void* d_out, int out_size, void* d_ws, size_t ws_size,
                              hipStream_t stream) {
    const float* g  = (const float*)d_in[0];
    const float* W1 = (const float*)d_in[1];
    const float* b1 = (const float*)d_in[2];
    const float* W2 = (const float*)d_in[3];
    const float* b2 = (const float*)d_in[4];
    const float* W3 = (const float*)d_in[5];
    const float* b3 = (const float*)d_in[6];
    float* out = (float*)d_out;

    const int A = in_sizes[2] / NH;          // b1 is [A, 32] -> 1000
    const int S = in_sizes[0] / (A * NI);    // g is [S, A, 5] -> 2000

    grouped_mlp_wmma<<<A, 256, 0, stream>>>(g, W1, b1, W2, b2, W3, b3, out, S, A);
}